// deeperNet_14224931685024
// MI455X (gfx1250) — compile-verified
//
#include <hip/hip_runtime.h>

typedef __attribute__((ext_vector_type(2))) float v2f;
typedef __attribute__((ext_vector_type(8))) float v8f;

#define LEAKY_SLOPE 0.01f

__device__ __forceinline__ void atomAddF(float* p, float v) {
    __hip_atomic_fetch_add(p, v, __ATOMIC_RELAXED, __HIP_MEMORY_SCOPE_AGENT);
}

// ---------------- zero fill (float4 granularity) ----------------
__global__ void k_zero4(float4* __restrict__ p, int n4) {
    int i = blockIdx.x * blockDim.x + threadIdx.x;
    if (i < n4) p[i] = make_float4(0.f, 0.f, 0.f, 0.f);
}

// ---------------- degree counting ----------------
__global__ void k_degrees(const int* __restrict__ src, const int* __restrict__ dst,
                          float* __restrict__ degO, float* __restrict__ degI, int E) {
    int e = blockIdx.x * blockDim.x + threadIdx.x;
    if (e >= E) return;
    atomAddF(&degO[src[e]], 1.0f);
    atomAddF(&degI[dst[e]], 1.0f);
}

// deg -> rsqrt(max(deg,1)) in place, both arrays
__global__ void k_inv_sqrt(float* __restrict__ a, float* __restrict__ b, int n) {
    int i = blockIdx.x * blockDim.x + threadIdx.x;
    if (i >= n) return;
    a[i] = rsqrtf(fmaxf(a[i], 1.0f));
    b[i] = rsqrtf(fmaxf(b[i], 1.0f));
}

// ---------------- edge-parallel gather + scatter-add ----------------
// agg[dst] += x[src] * inv_sqrt_out[src]; one float4 chunk per thread
__global__ void k_gather(const float* __restrict__ x, const float* __restrict__ invO,
                         const int* __restrict__ src, const int* __restrict__ dst,
                         float* __restrict__ agg, int E, int lgChunks, int d) {
    int tid = blockIdx.x * blockDim.x + threadIdx.x;
    int e = tid >> lgChunks;
    if (e >= E) return;
    int c = tid & ((1 << lgChunks) - 1);
    int s = src[e];
    int t = dst[e];
    float w = invO[s];
    const float4 v = *reinterpret_cast<const float4*>(x + (size_t)s * d + c * 4);
    float* base = agg + (size_t)t * d + c * 4;
    atomAddF(base + 0, v.x * w);
    atomAddF(base + 1, v.y * w);
    atomAddF(base + 2, v.z * w);
    atomAddF(base + 3, v.w * w);
}

// ---------------- WMMA fp32 GEMM: out = act((agg * invI) @ W + b) ----------------
// One wave computes a 64-row x 16-col block (4 sub-tiles of 16x16).
// B fragments for all K-steps preloaded once into registers and reused.
// Wave id scalarized via readfirstlane so tile indices/guards live in SGPRs.
// V_WMMA_F32_16X16X4_F32 layouts:
//   A 16x4: lanes 0-15 -> M=lane, K={kk,kk+1}; lanes 16-31 -> M=lane-16, K={kk+2,kk+3}
//   B 4x16 (mirror): N=lane&15; lanes<16 K={kk,kk+1}, lanes>=16 K={kk+2,kk+3}
//   C/D: VGPR j -> row j (lanes<16) / row j+8 (lanes>=16), col=lane&15
template <int KDIM, int MD, bool LEAKY>
__global__ void k_gemm_wmma(const float* __restrict__ agg, const float* __restrict__ invI,
                            const float* __restrict__ W, const float* __restrict__ bias,
                            float* __restrict__ out, int Nn, int colTiles, int rowBlocks) {
    constexpr int NK = KDIM / 4;
    constexpr bool FULLCOL = (MD % 16 == 0);

    // wave-uniform tile coordinates, forced into SGPRs
    int gwave    = __builtin_amdgcn_readfirstlane(
                       (int)((blockIdx.x * blockDim.x + threadIdx.x) >> 5));
    int lane     = threadIdx.x & 31;
    int rowBlock = gwave / colTiles;
    int colTile  = gwave - rowBlock * colTiles;
    if (rowBlock >= rowBlocks) return;   // scalar-branch exit

    int   ln15    = lane & 15;
    int   col     = colTile * 16 + ln15;
    int   colC    = FULLCOL ? col : ((col < MD) ? col : (MD - 1));
    float colMask = FULLCOL ? 1.f : ((col < MD) ? 1.f : 0.f);
    int   kHalf   = (lane >> 4) * 2;     // 0 for lanes 0-15, 2 for 16-31
    int   hi      = lane >> 4;

    // preload all B fragments (registers, fully unrolled, branch-free)
    v2f bf[NK];
    #pragma unroll
    for (int t = 0; t < NK; ++t) {
        int k0 = t * 4 + kHalf;
        bf[t].x = W[k0 * MD + colC] * colMask;
        bf[t].y = W[(k0 + 1) * MD + colC] * colMask;
    }
    float bv = bias[colC] * colMask;

    #pragma unroll
    for (int st = 0; st < 4; ++st) {
        int tileBase = rowBlock * 64 + st * 16;   // SGPR (scalar guard below)
        if (tileBase >= Nn) break;
        int   mrow  = tileBase + ln15;
        int   mrowC = (mrow < Nn) ? mrow : (Nn - 1);
        float s     = invI[mrowC];

        v8f acc = {0.f, 0.f, 0.f, 0.f, 0.f, 0.f, 0.f, 0.f};
        #pragma unroll
        for (int t = 0; t < NK; ++t) {
            int k0 = t * 4 + kHalf;
            v2f a = *reinterpret_cast<const v2f*>(agg + (size_t)mrowC * KDIM + k0);
            a.x *= s;
            a.y *= s;
            acc = __builtin_amdgcn_wmma_f32_16x16x4_f32(
                false, a, false, bf[t], (short)0, acc, false, false);
        }

        // activation: leaky(x) == max(x, slope*x) for 0<slope<1 (2 VALU ops)
        float h[8];
        #pragma unroll
        for (int j = 0; j < 8; ++j) {
            float v = acc[j] + bv;
            h[j] = LEAKY ? fmaxf(v, LEAKY_SLOPE * v) : v;
        }

        // epilogue: this lane owns rows tileBase+hi*8+j, column `col`
        float* outBase = out + (size_t)(tileBase + hi * 8) * MD + col;
        if (FULLCOL && (tileBase + 16 <= Nn)) {
            // fast path: full tile, unconditional coalesced clause stores
            #pragma unroll
            for (int j = 0; j < 8; ++j) outBase[j * MD] = h[j];
        } else {
            #pragma unroll
            for (int j = 0; j < 8; ++j) {
                int r = tileBase + hi * 8 + j;
                if (col < MD && r < Nn) outBase[j * MD] = h[j];
            }
        }
    }
}

extern "C" void kernel_launch(void* const* d_in, const int* in_sizes, int n_in,
                              void* d_out, int out_size, void* d_ws, size_t ws_size,
                              hipStream_t stream) {
    (void)n_in; (void)out_size; (void)ws_size;

    const float* features = (const float*)d_in[0];
    const int*   esrc     = (const int*)d_in[1];
    const int*   edst     = (const int*)d_in[2];

    const int N = in_sizes[0] / 4;   // 100000
    const int E = in_sizes[1];       // 3200000
    static const int DIMS[9] = {4, 64, 64, 64, 64, 64, 64, 64, 3};

    // workspace layout (float offsets, aligned)
    float* ws      = (float*)d_ws;
    float* inv_out = ws;                         // N floats (deg_out -> rsqrt)
    float* inv_in  = ws + 100352;                // N floats (deg_in  -> rsqrt)
    float* xbuf    = ws + 200704;                // N*64 activations
    float* agg     = ws + 200704 + 6400000;      // N*64 aggregation

    const int BT = 256;

    // 1) degrees -> inv sqrt
    {
        int n4 = (N + 3) / 4;
        k_zero4<<<(n4 + BT - 1) / BT, BT, 0, stream>>>((float4*)inv_out, n4);
        k_zero4<<<(n4 + BT - 1) / BT, BT, 0, stream>>>((float4*)inv_in,  n4);
        k_degrees<<<(E + BT - 1) / BT, BT, 0, stream>>>(esrc, edst, inv_out, inv_in, E);
        k_inv_sqrt<<<(N + BT - 1) / BT, BT, 0, stream>>>(inv_out, inv_in, N);
    }

    // 2) 8 GCN layers
    const float* xcur = features;
    for (int i = 0; i < 8; ++i) {
        const int Kd = DIMS[i];
        const int Md = DIMS[i + 1];
        const float* Wi = (const float*)d_in[3 + 2 * i];
        const float* bi = (const float*)d_in[4 + 2 * i];

        // zero agg
        int n4 = N * Kd / 4;
        k_zero4<<<(n4 + BT - 1) / BT, BT, 0, stream>>>((float4*)agg, n4);

        // gather + scatter-add
        int chunks = Kd / 4;                       // 1 or 16
        int lg = (chunks == 16) ? 4 : 0;
        long long total = (long long)E * chunks;
        int blocks = (int)((total + BT - 1) / BT);
        k_gather<<<blocks, BT, 0, stream>>>(xcur, inv_out, esrc, edst, agg, E, lg, Kd);

        // WMMA GEMM + bias (+ leaky, except last layer)
        int colTiles  = (Md + 15) / 16;
        int rowBlocks = (N + 63) / 64;
        long long waves = (long long)rowBlocks * colTiles;
        long long thr = waves * 32;
        int gblocks = (int)((thr + BT - 1) / BT);
        float* yout = (i == 7) ? (float*)d_out : xbuf;

        if (i == 0) {
            k_gemm_wmma<4, 64, true><<<gblocks, BT, 0, stream>>>(
                agg, inv_in, Wi, bi, yout, N, colTiles, rowBlocks);
        } else if (i < 7) {
            k_gemm_wmma<64, 64, true><<<gblocks, BT, 0, stream>>>(
                agg, inv_in, Wi, bi, yout, N, colTiles, rowBlocks);
        } else {
            k_gemm_wmma<64, 3, false><<<gblocks, BT, 0, stream>>>(
                agg, inv_in, Wi, bi, yout, N, colTiles, rowBlocks);
        }
        xcur = xbuf;
    }
}